// Attention_25262997635932
// MI455X (gfx1250) — compile-verified
//
#include <hip/hip_runtime.h>
#include <hip/hip_bf16.h>
#include <math.h>

#define HD 4096   // hidden size
#define SL 4096   // seq len
#define BT 2      // batch

typedef __attribute__((ext_vector_type(2))) float v2f;
typedef __attribute__((ext_vector_type(8))) float v8f;

// ---------------------------------------------------------------------------
// Kernel 0: cbias[b] = dot(bias, hidden[b])   (tiny, one workgroup)
// ---------------------------------------------------------------------------
__global__ void bias_dot_kernel(const float* __restrict__ hidden,
                                const float* __restrict__ bias,
                                float* __restrict__ cbias) {
    __shared__ float red0[256];
    __shared__ float red1[256];
    const int t = threadIdx.x;
    float a0 = 0.f, a1 = 0.f;
    for (int i = t; i < HD; i += 256) {
        const float bb = bias[i];
        a0 += bb * hidden[i];
        a1 += bb * hidden[HD + i];
    }
    red0[t] = a0; red1[t] = a1;
    __syncthreads();
    for (int s = 128; s > 0; s >>= 1) {
        if (t < s) { red0[t] += red0[t + s]; red1[t] += red1[t + s]; }
        __syncthreads();
    }
    if (t == 0) { cbias[0] = red0[0]; cbias[1] = red1[0]; }
}

// ---------------------------------------------------------------------------
// Kernel 1: v[b, :] = hidden[b, :] @ W   via V_WMMA_F32_16X16X4_F32
//   M = 2 (padded to 16), K = N = 4096.
//   Grid: 256 blocks (one 16-col tile each) x 8 waves (K-slice 512 each).
//   A 16x4 f32 tile: lane m=lane&15 holds row m; lanes 0-15 -> K={0,1},
//   lanes 16-31 -> K={2,3} (VGPR0 = even K of pair, VGPR1 = odd).
//   B 4x16 f32 tile: lane col=lane&15 holds col; same K split across halves.
//   C tile: VGPR r, lanes 0-15 = row M=r, col=lane.  We only need rows 0,1.
// ---------------------------------------------------------------------------
__global__ void hidden_w_gemv_wmma(const float* __restrict__ hidden,
                                   const float* __restrict__ W,
                                   float* __restrict__ v) {
    __shared__ float part[8][2][16];

    const int wave = threadIdx.x >> 5;
    const int lane = threadIdx.x & 31;
    const int n0   = blockIdx.x * 16;
    const int m    = lane & 15;            // A row / B col index
    const int koff = (lane >> 4) * 2;      // K sub-offset per lane half
    const int k0   = wave * 512;           // this wave's K slice

    // branch-free zero padding of A rows >= 2 (keeps EXEC uniform)
    const float mscale = (m < BT) ? 1.f : 0.f;
    const float* __restrict__ arow = hidden + (size_t)(m & 1) * HD;

    v8f acc = {};
    for (int k = k0; k < k0 + 512; k += 4) {
        v2f a, b;
        a.x = arow[k + koff]     * mscale;
        a.y = arow[k + koff + 1] * mscale;
        const float* __restrict__ wp = W + (size_t)(k + koff) * HD + n0 + m;
        b.x = wp[0];
        b.y = wp[HD];
        acc = __builtin_amdgcn_wmma_f32_16x16x4_f32(
            /*neg_a=*/false, a, /*neg_b=*/false, b,
            /*c_mod=*/(short)0, acc, /*reuse_a=*/false, /*reuse_b=*/false);
    }

    if ((lane >> 4) == 0) {                // lanes 0-15 hold rows M=0..7
        part[wave][0][m] = acc[0];         // row M=0
        part[wave][1][m] = acc[1];         // row M=1
    }
    __syncthreads();

    if (threadIdx.x < 32) {
        const int row = threadIdx.x >> 4;  // 0..1
        const int col = threadIdx.x & 15;
        float s = 0.f;
        #pragma unroll
        for (int wv = 0; wv < 8; ++wv) s += part[wv][row][col];
        v[(size_t)row * HD + n0 + col] = s;
    }
}

// ---------------------------------------------------------------------------
// Kernel 2 (fused scores + softmax): one wave per timestep s.
//   scores[s,b] = enc[s,b,:] . v[b,:] + cbias[b];  softmax over b (size 2).
//   Streams 128 MB of enc once with float4 (global_load_b128); v (32 KB) is
//   staged in LDS once per workgroup so 8 waves share it (ds_load_b128),
//   keeping the global-load path dedicated to the HBM stream.
// ---------------------------------------------------------------------------
__global__ void scores_softmax_kernel(const float* __restrict__ enc,
                                      const float* __restrict__ v,
                                      const float* __restrict__ cbias,
                                      float* __restrict__ out) {
    __shared__ float4 sv[2 * HD / 4];      // 32 KB: v[0,:] then v[1,:]

    const int tid  = threadIdx.x;
    const int wave = tid >> 5;
    const int lane = tid & 31;
    const int s    = blockIdx.x * 8 + wave;

    // cooperative stage of v into LDS (2048 float4 / 256 threads = 8 each)
    {
        const float4* __restrict__ vg = (const float4*)v;
        #pragma unroll
        for (int i = 0; i < 8; ++i)
            sv[tid + i * 256] = vg[tid + i * 256];
    }
    __syncthreads();

    const float4* __restrict__ e0 = (const float4*)(enc + (size_t)s * (BT * HD));
    const float4* __restrict__ e1 = e0 + (HD / 4);
    const float4* __restrict__ v0 = sv;
    const float4* __restrict__ v1 = sv + (HD / 4);

    float acc0 = 0.f, acc1 = 0.f;
    #pragma unroll 4
    for (int i = lane; i < HD / 4; i += 32) {
        const float4 a  = e0[i];
        const float4 w0 = v0[i];
        acc0 += a.x * w0.x + a.y * w0.y + a.z * w0.z + a.w * w0.w;
        const float4 b  = e1[i];
        const float4 w1 = v1[i];
        acc1 += b.x * w1.x + b.y * w1.y + b.z * w1.z + b.w * w1.w;
    }
    #pragma unroll
    for (int off = 16; off > 0; off >>= 1) {
        acc0 += __shfl_down(acc0, off, 32);
        acc1 += __shfl_down(acc1, off, 32);
    }
    if (lane == 0) {
        const float s0 = acc0 + cbias[0];
        const float s1 = acc1 + cbias[1];
        const float mx = fmaxf(s0, s1);
        const float x0 = __expf(s0 - mx);
        const float x1 = __expf(s1 - mx);
        const float inv = 1.f / (x0 + x1);
        out[(size_t)s * 2 + 0] = x0 * inv;
        out[(size_t)s * 2 + 1] = x1 * inv;
    }
}

// ---------------------------------------------------------------------------
extern "C" void kernel_launch(void* const* d_in, const int* in_sizes, int n_in,
                              void* d_out, int out_size, void* d_ws, size_t ws_size,
                              hipStream_t stream) {
    const float* hidden = (const float*)d_in[0];   // [2, 1, 4096]
    const float* enc    = (const float*)d_in[1];   // [4096, 2, 4096]
    const float* W      = (const float*)d_in[2];   // [4096, 4096]
    const float* bias   = (const float*)d_in[3];   // [4096]
    float* out = (float*)d_out;                    // [1,1,4096,2] -> 8192 f32

    float* v     = (float*)d_ws;                   // 2*4096 floats
    float* cbias = v + 2 * HD;                     // 2 floats

    bias_dot_kernel<<<1, 256, 0, stream>>>(hidden, bias, cbias);
    hidden_w_gemv_wmma<<<HD / 16, 256, 0, stream>>>(hidden, W, v);
    scores_softmax_kernel<<<SL / 8, 256, 0, stream>>>(enc, v, cbias, out);
}